// WTLayer_21492016350023
// MI455X (gfx1250) — compile-verified
//
#include <hip/hip_runtime.h>

// ---------------------------------------------------------------------------
// 2-level DWT (db6, 12 taps) on (1024, 16384) fp32 rows.
// Roofline: 256 MB HBM traffic vs 0.6 GFLOP -> pure bandwidth problem (~11us
// at 23.3 TB/s). One workgroup per row; row staged in LDS (147.6 KB << 320 KB
// per WGP on CDNA5); both DWT levels computed from LDS via
// V_WMMA_F32_16X16X4_F32 (M=filter row, K=4-tap chunk x3, N=16 outputs);
// single interleaved float3 output pass (12 contiguous bytes per lane).
// Round 2: branchless symmetric-extension indexing (min/max instead of
// EXEC-mask branches) and all 6 B-operand LDS gathers hoisted ahead of the
// 3 back-to-back WMMAs per tile.
// ---------------------------------------------------------------------------

typedef __attribute__((ext_vector_type(2))) float v2f;
typedef __attribute__((ext_vector_type(8))) float v8f;

#define N1 16384   // input length per row
#define M1 8197    // level-1 output length = floor((N1+11)/2)
#define M2 4104    // level-2 output length = floor((M1+11)/2)
#define NT1 513    // ceil(M1/16)
#define NT2 257    // ceil(M2/16)

// kern row 0 = dec_lo reversed; kern row 1 = dec_hi reversed = d[i]*(-1)^i
__constant__ float KLO[12] = {
    0.11154074335008017f,  0.4946238903983854f,   0.7511339080215775f,
    0.3152503517092432f,  -0.22626469396516913f, -0.12976686756709563f,
    0.09750160558707936f,  0.02752286553001629f, -0.031582039318031156f,
    0.0005538422009938016f, 0.004777257511010651f, -0.00107730108499558f};
__constant__ float KHI[12] = {
   -0.00107730108499558f, -0.004777257511010651f, 0.0005538422009938016f,
    0.031582039318031156f, 0.02752286553001629f,  -0.09750160558707936f,
   -0.12976686756709563f,  0.22626469396516913f,   0.3152503517092432f,
   -0.7511339080215775f,   0.4946238903983854f,   -0.11154074335008017f};

// symmetric extension, left-trimmed by 1 (pad (11,11) then [1:]).
// ext[j] = sig[idx], j in [0, n+20].  t = j-10:
//   t < 0  -> -1-t  (= ~t; selected by max(t, ~t))
//   t >= n -> 2n-1-t (selected by min against reflection; valid for j<=2n+9)
__device__ __forceinline__ int sym_idx(int j, int n) {
  const int t  = j - 10;
  const int t2 = max(t, ~t);
  return min(t2, 2 * n - 1 - t2);
}

struct F3 { float x, y, z; };

__global__ __launch_bounds__(256) void wt_kernel(const float* __restrict__ x,
                                                 float* __restrict__ out) {
  __shared__ float s_sig[N1];
  __shared__ float s_cA1[M1];
  __shared__ float s_cD1[M1];
  __shared__ float s_cD2[M2];

  const int tid  = threadIdx.x;
  const int lane = tid & 31;
  const int wave = tid >> 5;
  const int b    = blockIdx.x;

  // ---- stage 1: row -> LDS (float4, fully coalesced) ----
  {
    const float4* in4 = (const float4*)(x + (size_t)b * N1);
    float4* s4 = (float4*)s_sig;
#pragma unroll
    for (int j = 0; j < N1 / 4 / 256; ++j) s4[tid + 256 * j] = in4[tid + 256 * j];
  }
  __syncthreads();

  // ---- A operand: constant per lane. A is 16x4 f32: lanes 0-15 hold K=0,1
  // (a.x,a.y), lanes 16-31 hold K=2,3. Row m: 0 -> lo filter, 1 -> hi, else 0.
  const int m  = lane & 15;
  const int kh = (lane < 16) ? 0 : 2;
  v2f a[3];
#pragma unroll
  for (int c = 0; c < 3; ++c) {
    const int i0 = 4 * c + kh;
    float f0 = 0.0f, f1 = 0.0f;
    if (m == 0)      { f0 = KLO[i0]; f1 = KLO[i0 + 1]; }
    else if (m == 1) { f0 = KHI[i0]; f1 = KHI[i0 + 1]; }
    a[c].x = f0; a[c].y = f1;
  }

  // ---- stage 2: level-1 DWT: 513 tiles of 16 outputs, 3 WMMAs each ----
  for (int t = wave; t < NT1; t += 8) {
    const int k0 = t << 4;
    const int jb0 = 2 * (k0 + m) + kh;   // B gather base for this lane
    v8f acc = {0.f, 0.f, 0.f, 0.f, 0.f, 0.f, 0.f, 0.f};
#if __has_builtin(__builtin_amdgcn_wmma_f32_16x16x4_f32)
    v2f bv[3];
#pragma unroll
    for (int c = 0; c < 3; ++c) {        // issue all 6 LDS gathers first
      bv[c].x = s_sig[sym_idx(jb0 + 4 * c,     N1)];
      bv[c].y = s_sig[sym_idx(jb0 + 4 * c + 1, N1)];
    }
#pragma unroll
    for (int c = 0; c < 3; ++c)          // 3 back-to-back WMMAs
      acc = __builtin_amdgcn_wmma_f32_16x16x4_f32(false, a[c], false, bv[c],
                                                  (short)0, acc, false, false);
#else
    {
      float ylo = 0.f, yhi = 0.f;
      const int k = k0 + m;
#pragma unroll
      for (int i = 0; i < 12; ++i) {
        const float v = s_sig[sym_idx(2 * k + i, N1)];
        ylo = fmaf(KLO[i], v, ylo);
        yhi = fmaf(KHI[i], v, yhi);
      }
      acc[0] = ylo; acc[1] = yhi;
    }
#endif
    // D layout: VGPR0 lanes 0-15 = row M=0 (lo), VGPR1 lanes 0-15 = row M=1 (hi)
    const int k = k0 + m;
    if (lane < 16 && k < M1) { s_cA1[k] = acc[0]; s_cD1[k] = acc[1]; }
  }
  __syncthreads();

  // ---- stage 3: level-2 DWT on cA1 (keep only the hi band) ----
  for (int t = wave; t < NT2; t += 8) {
    const int k0 = t << 4;
    const int jb0 = 2 * (k0 + m) + kh;
    v8f acc = {0.f, 0.f, 0.f, 0.f, 0.f, 0.f, 0.f, 0.f};
#if __has_builtin(__builtin_amdgcn_wmma_f32_16x16x4_f32)
    v2f bv[3];
#pragma unroll
    for (int c = 0; c < 3; ++c) {
      bv[c].x = s_cA1[sym_idx(jb0 + 4 * c,     M1)];
      bv[c].y = s_cA1[sym_idx(jb0 + 4 * c + 1, M1)];
    }
#pragma unroll
    for (int c = 0; c < 3; ++c)
      acc = __builtin_amdgcn_wmma_f32_16x16x4_f32(false, a[c], false, bv[c],
                                                  (short)0, acc, false, false);
#else
    {
      float yhi = 0.f;
      const int k = k0 + m;
#pragma unroll
      for (int i = 0; i < 12; ++i)
        yhi = fmaf(KHI[i], s_cA1[sym_idx(2 * k + i, M1)], yhi);
      acc[1] = yhi;
    }
#endif
    const int k = k0 + m;
    if (lane < 16 && k < M2) s_cD2[k] = acc[1];
  }
  __syncthreads();

  // ---- stage 4: interleaved output (b, n, 3) = {sig, cD2|0, cD1|0} ----
  {
    F3* op3 = (F3*)(out + (size_t)b * N1 * 3);
#pragma unroll 4
    for (int j = 0; j < N1 / 256; ++j) {
      const int n = tid + 256 * j;
      F3 v;
      v.x = s_sig[n];
      v.y = (n < M2) ? s_cD2[n] : 0.0f;
      v.z = (n < M1) ? s_cD1[n] : 0.0f;
      op3[n] = v;
    }
  }
}

extern "C" void kernel_launch(void* const* d_in, const int* in_sizes, int n_in,
                              void* d_out, int out_size, void* d_ws, size_t ws_size,
                              hipStream_t stream) {
  (void)n_in; (void)out_size; (void)d_ws; (void)ws_size;
  const float* x = (const float*)d_in[0];
  float* out = (float*)d_out;
  const int B = in_sizes[0] / N1;  // 1024 rows
  wt_kernel<<<dim3(B), dim3(256), 0, stream>>>(x, out);
}